// MPLayer_37039797960981
// MI455X (gfx1250) — compile-verified
//
#include <hip/hip_runtime.h>

typedef __attribute__((ext_vector_type(2))) float v2f;
typedef __attribute__((ext_vector_type(4))) float v4f;
typedef __attribute__((ext_vector_type(8))) float v8f;

#define DH 5            // node feature dim
#define DE 11           // edge feature dim
#define DEG 16          // neighbors per node
#define XIN 21          // 2*DH + DE
#define KPAD 24         // K padded to multiple of 4 for wmma 16x16x4
#define XSTRIDE 26      // LDS row stride (floats): conflict-free, 8B aligned rows
#define WAVES 8
#define NODES_PER_BLOCK (WAVES * 16)

__global__ __launch_bounds__(256) void mp_step_kernel(
    const float* __restrict__ h_src, const int* __restrict__ nbrs,
    const float* __restrict__ ef,
    const float* __restrict__ Vw, const float* __restrict__ Vb,
    const float* __restrict__ Ew, const float* __restrict__ Eb,
    const float* __restrict__ Uw, const float* __restrict__ Ub,
    float* __restrict__ h_dst, int n_nodes, int k)
{
    __shared__ float Xs[WAVES][16][XSTRIDE];   // per-wave 16x24 input tile (padded)
    __shared__ float Wc[KPAD * DH];            // combined 24x5 weight (rows 21..23 = 0)
    __shared__ float Bc[DH];                   // combined bias

    const int t = threadIdx.x;

    // ---- build combined weight W_comb = [U(0:5); V@U(5:10); E@U(10:21); 0] ----
    if (t < KPAD * DH) {
        int r = t % KPAD, c = t / KPAD;        // r: input row 0..23, c: out col 0..4
        float acc = 0.f;
        if (r < DH) {
            acc = Uw[r * DH + c];
        } else if (r < 2 * DH) {
            int i = r - DH;
            for (int j = 0; j < DH; ++j) acc += Vw[i * DH + j] * Uw[(DH + j) * DH + c];
        } else if (r < XIN) {
            int i = r - 2 * DH;
            for (int j = 0; j < DE; ++j) acc += Ew[i * DE + j] * Uw[(2 * DH + j) * DH + c];
        }
        Wc[r * DH + c] = acc;
    } else if (t < KPAD * DH + DH) {
        int c = t - KPAD * DH;
        float acc = Ub[c];
        for (int j = 0; j < DH; ++j) acc += Vb[j] * Uw[(DH + j) * DH + c];
        for (int j = 0; j < DE; ++j) acc += Eb[j] * Uw[(2 * DH + j) * DH + c];
        Bc[c] = acc;
    }
    __syncthreads();

    const int wave = t >> 5;
    const int lane = t & 31;
    const int half = lane >> 4;   // 0: lanes 0-15, 1: lanes 16-31
    const int lr   = lane & 15;   // A-matrix row / D-matrix column
    const int tileBase = blockIdx.x * NODES_PER_BLOCK + wave * 16;
    const int node  = tileBase + lr;
    const bool valid = node < n_nodes;
    const int  nsafe = valid ? node : 0;

    // ---- stage X = [h_own(5) | h_nbr(5) | edge(11) | 0 0 0] into LDS ----
    if (half == 0) {
        int nb = nbrs[(size_t)nsafe * DEG + k];
        #pragma unroll
        for (int j = 0; j < DH; ++j)
            Xs[wave][lr][j] = valid ? h_src[(size_t)nsafe * DH + j] : 0.f;
        #pragma unroll
        for (int j = 0; j < DH; ++j)
            Xs[wave][lr][DH + j] = valid ? h_src[(size_t)nb * DH + j] : 0.f;
        Xs[wave][lr][21] = 0.f; Xs[wave][lr][22] = 0.f; Xs[wave][lr][23] = 0.f;
    } else {
        size_t eb = ((size_t)nsafe * DEG + k) * DE;
        #pragma unroll
        for (int j = 0; j < DE; ++j)
            Xs[wave][lr][2 * DH + j] = valid ? ef[eb + j] : 0.f;
    }
    __syncthreads();

    // ---- B operand (W_comb columns striped across lanes; cols >=5 are zero) ----
    v2f b[6];
    #pragma unroll
    for (int ch = 0; ch < 6; ++ch) {
        int r0 = 4 * ch + 2 * half;
        if (lr < DH) { b[ch].x = Wc[r0 * DH + lr]; b[ch].y = Wc[(r0 + 1) * DH + lr]; }
        else         { b[ch].x = 0.f;              b[ch].y = 0.f; }
    }

    // ---- C seeded with bias (per output column = lr) ----
    float binit = (lr < DH) ? Bc[lr] : 0.f;
    v8f c;
    #pragma unroll
    for (int v = 0; v < 8; ++v) c[v] = binit;

    // ---- 6 chained f32 WMMAs: D = X(16x24) @ W(24x16) + C ----
    #pragma unroll
    for (int ch = 0; ch < 6; ++ch) {
        v2f a;
        a.x = Xs[wave][lr][4 * ch + 2 * half];
        a.y = Xs[wave][lr][4 * ch + 2 * half + 1];
        c = __builtin_amdgcn_wmma_f32_16x16x4_f32(
                /*neg_a=*/false, a, /*neg_b=*/false, b[ch],
                /*c_mod=*/(short)0, c, /*reuse_a=*/false, /*reuse_b=*/false);
    }

    // ---- epilogue: stage D (16 nodes x 5) in LDS, store coalesced ----
    // Per-wave output block is 80 contiguous floats at h_dst + tileBase*5.
    float* ds = &Xs[wave][0][0];               // >= 80 floats of scratch per wave
    if (lr < DH) {
        #pragma unroll
        for (int v = 0; v < 8; ++v)
            ds[(v + 8 * half) * DH + lr] = c[v];
    }
    __builtin_amdgcn_wave_barrier();           // keep DS stores before DS reads

    const size_t outBase = (size_t)tileBase * DH;
    if (tileBase + 16 <= n_nodes) {
        // full tile: lanes 0..19 each store one 16B chunk (base is 16B aligned)
        if (lane < 20) {
            v4f val;
            val.x = ds[lane * 4 + 0];
            val.y = ds[lane * 4 + 1];
            val.z = ds[lane * 4 + 2];
            val.w = ds[lane * 4 + 3];
            *(v4f*)(h_dst + outBase + lane * 4) = val;
        }
    } else if (tileBase < n_nodes) {
        // partial tail tile: masked scalar stores
        int nvalid = (n_nodes - tileBase) * DH; // valid floats in this tile
        for (int i = lane; i < 80; i += 32)
            if (i < nvalid) h_dst[outBase + i] = ds[i];
    }
}

extern "C" void kernel_launch(void* const* d_in, const int* in_sizes, int n_in,
                              void* d_out, int out_size, void* d_ws, size_t ws_size,
                              hipStream_t stream) {
    const float* h0  = (const float*)d_in[0];
    const int*   nbr = (const int*)d_in[1];
    const float* ef  = (const float*)d_in[2];
    const float* Vw  = (const float*)d_in[3];
    const float* Vb  = (const float*)d_in[4];
    const float* Ew  = (const float*)d_in[5];
    const float* Eb  = (const float*)d_in[6];
    const float* Uw  = (const float*)d_in[7];
    const float* Ub  = (const float*)d_in[8];
    float* out = (float*)d_out;

    const int n_nodes = in_sizes[0] / DH;
    float* buf0 = (float*)d_ws;
    float* buf1 = buf0 + (size_t)n_nodes * DH;

    const int grid = (n_nodes + NODES_PER_BLOCK - 1) / NODES_PER_BLOCK;

    // 16 globally-synchronized steps (kernel-launch boundaries), ping-pong h.
    for (int k = 0; k < DEG; ++k) {
        const float* src = (k == 0) ? h0 : ((k % 2 == 1) ? buf0 : buf1);
        float* dst = (k == DEG - 1) ? out : ((k % 2 == 0) ? buf0 : buf1);
        mp_step_kernel<<<grid, 256, 0, stream>>>(src, nbr, ef, Vw, Vb, Ew, Eb,
                                                 Uw, Ub, dst, n_nodes, k);
    }
}